// MultiDimHiPPO_89103391523690
// MI455X (gfx1250) — compile-verified
//
#include <hip/hip_runtime.h>
#include <hip/hip_bf16.h>

typedef __attribute__((ext_vector_type(2))) float v2f;
typedef __attribute__((ext_vector_type(8))) float v8f;

// ---------------------------------------------------------------------------
// Kernel 1: build V[t, :] = dA^{1023-t} @ dB  via log-doubling (10 stages).
// Single workgroup of 256 threads. P = dA^s kept in LDS (stride 65 to avoid
// 64-way bank conflicts on column reads). V lives in global workspace.
// ---------------------------------------------------------------------------
__global__ void __launch_bounds__(256) hippo_build_v(
    const float* __restrict__ dA,   // (64,64) row-major
    const float* __restrict__ dB,   // (64,)
    float* __restrict__ V)          // (1024,64) row-major, workspace
{
    __shared__ float P[64 * 65];
    __shared__ float Q[64 * 65];
    const int tid = threadIdx.x;

    // P = dA
    for (int i = tid; i < 4096; i += 256) {
        P[(i >> 6) * 65 + (i & 63)] = dA[i];
    }
    // V[1023] = dB  (dA^0 term)
    if (tid < 64) V[1023 * 64 + tid] = dB[tid];
    __threadfence();
    __syncthreads();

    int s = 1;
    for (int stage = 0; stage < 10; ++stage) {
        // New rows t in [1024-2s, 1024-s):  V[t] = P @ V[t+s]
        const int base = 1024 - 2 * s;
        const int total = s * 64;
        for (int idx = tid; idx < total; idx += 256) {
            const int tl = idx >> 6;   // local row
            const int m  = idx & 63;
            const float* __restrict__ vin = V + (size_t)(base + tl + s) * 64;
            float sum = 0.0f;
#pragma unroll 8
            for (int n = 0; n < 64; ++n)
                sum = __builtin_fmaf(P[m * 65 + n], vin[n], sum);
            V[(size_t)(base + tl) * 64 + m] = sum;
        }

        if (stage < 9) {
            // Q = P @ P  (then P <- Q) : advance to dA^{2s}
            for (int idx = tid; idx < 4096; idx += 256) {
                const int r = idx >> 6, c = idx & 63;
                float sum = 0.0f;
#pragma unroll 8
                for (int n = 0; n < 64; ++n)
                    sum = __builtin_fmaf(P[r * 65 + n], P[n * 65 + c], sum);
                Q[r * 65 + c] = sum;
            }
            __syncthreads();   // all reads of P (expand + square) done
            for (int idx = tid; idx < 4096; idx += 256) {
                const int r = idx >> 6, c = idx & 63;
                P[r * 65 + c] = Q[r * 65 + c];
            }
        }
        __threadfence();       // make new V rows visible for next stage
        __syncthreads();
        s <<= 1;
    }
}

// ---------------------------------------------------------------------------
// Kernel 2: per batch b, C_b (64x64) = X_b^T (64x1024) @ V (1024x64),
// out[b*4096 + d*64 + m] = C_b[d, m].
// Grid = 32 blocks (one per batch), 512 threads = 16 wave32's.
// Each wave owns one 16x16 output tile, K accumulated with
// v_wmma_f32_16x16x4_f32 (fp32 in, fp32 accumulate — matches reference).
// Main K loop (k < 992) carries an unconditional near-scope prefetch 16 rows
// ahead; an 8-iteration tail runs without prefetch so the hot loop has zero
// per-iteration branching.
// ---------------------------------------------------------------------------
__global__ void __launch_bounds__(512) hippo_gemm(
    const float* __restrict__ X,    // (32,1024,64) base_input
    const float* __restrict__ V,    // (1024,64)
    float* __restrict__ out)        // (1,32,4096)
{
    const int b    = blockIdx.x;
    const int wave = threadIdx.x >> 5;         // 0..15
    const int lane = threadIdx.x & 31;
    const int mi   = (wave >> 2) << 4;         // d-tile base  {0,16,32,48}
    const int ni   = (wave & 3) << 4;          // m-tile base  {0,16,32,48}
    const int half = lane >> 4;                // 0 = lanes 0-15, 1 = 16-31
    const int l16  = lane & 15;

    const float* __restrict__ Xb = X + (size_t)b * 1024 * 64;

    v8f acc = {};

    // Main loop: 248 iterations, unconditional prefetch 16 rows (4 KB) ahead.
#pragma unroll 8
    for (int k = 0; k < 992; k += 4) {
        // A-frag (16x4, M=d, K=t):  VGPR0 -> K = 2*half, VGPR1 -> K = 2*half+1
        // A[M,K] = Xb[K, M]  (coalesced: consecutive lanes, consecutive cols)
        const int kr = k + 2 * half;
        v2f a;
        a.x = Xb[(size_t)(kr + 0) * 64 + mi + l16];
        a.y = Xb[(size_t)(kr + 1) * 64 + mi + l16];

        // B-frag (4x16): VGPR i holds rows K=i (lanes 0-15) and K=i+2 (16-31)
        v2f bb;
        bb.x = V[(size_t)(kr + 0) * 64 + ni + l16];
        bb.y = V[(size_t)(kr + 1) * 64 + ni + l16];

        __builtin_prefetch(&Xb[(size_t)(k + 16) * 64 + mi + l16], 0, 3);

        acc = __builtin_amdgcn_wmma_f32_16x16x4_f32(
            /*neg_a=*/false, a, /*neg_b=*/false, bb,
            /*c_mod=*/(short)0, acc, /*reuse_a=*/false, /*reuse_b=*/false);
    }

    // Tail: last 8 K-steps, no prefetch.
#pragma unroll
    for (int k = 992; k < 1024; k += 4) {
        const int kr = k + 2 * half;
        v2f a;
        a.x = Xb[(size_t)(kr + 0) * 64 + mi + l16];
        a.y = Xb[(size_t)(kr + 1) * 64 + mi + l16];
        v2f bb;
        bb.x = V[(size_t)(kr + 0) * 64 + ni + l16];
        bb.y = V[(size_t)(kr + 1) * 64 + ni + l16];
        acc = __builtin_amdgcn_wmma_f32_16x16x4_f32(
            false, a, false, bb, (short)0, acc, false, false);
    }

    // C layout: VGPR i -> row M=i (lanes 0-15), row M=i+8 (lanes 16-31)
    float* __restrict__ ob = out + (size_t)b * 4096;
#pragma unroll
    for (int i = 0; i < 8; ++i) {
        const int d = mi + i + half * 8;
        const int m = ni + l16;
        ob[(size_t)d * 64 + m] = acc[i];
    }
}

extern "C" void kernel_launch(void* const* d_in, const int* in_sizes, int n_in,
                              void* d_out, int out_size, void* d_ws, size_t ws_size,
                              hipStream_t stream) {
    const float* X  = (const float*)d_in[0];   // base_input (32,1024,64) f32
    const float* dA = (const float*)d_in[1];   // (64,64) f32
    const float* dB = (const float*)d_in[2];   // (64,)  f32
    float* out = (float*)d_out;                // 131072 f32
    float* V   = (float*)d_ws;                 // needs 1024*64*4 = 256 KB

    hippo_build_v<<<1, 256, 0, stream>>>(dA, dB, V);
    hippo_gemm<<<32, 512, 0, stream>>>(X, V, out);
}